// EdgeLayer_13134009991287
// MI455X (gfx1250) — compile-verified
//
#include <hip/hip_runtime.h>
#include <math.h>

#define NN 40000
#define NE 640000
#define HD 128
#define BN_EPS 1e-5f

typedef __attribute__((ext_vector_type(2))) float v2f;
typedef __attribute__((ext_vector_type(8))) float v8f;

// ---- monotone float<->uint encoding so atomicMax(uint) == float max ----
__device__ __forceinline__ unsigned enc_f32(float f) {
    unsigned u = __float_as_uint(f);
    return (u & 0x80000000u) ? ~u : (u | 0x80000000u);
}
__device__ __forceinline__ float dec_f32(unsigned u) {
    return (u & 0x80000000u) ? __uint_as_float(u & 0x7FFFFFFFu)
                             : __uint_as_float(~u);
}
// enc(-inf) = ~0xFF800000 = 0x007FFFFF
#define ENC_NEG_INF 0x007FFFFFu

// ---------------- 0: init workspace (poisoned 0xAA between harness phases) --
__global__ void k_init(unsigned* __restrict__ smax, float* __restrict__ denom,
                       float* __restrict__ neigh, float* __restrict__ stats) {
    int i = blockIdx.x * blockDim.x + threadIdx.x;
    if (i < NN) { smax[i] = ENC_NEG_INF; denom[i] = 0.0f; }
    if (i < NN * HD) neigh[i] = 0.0f;
    if (i < 256) stats[i] = 0.0f;   // colsum[128] + colsumsq[128]
}

// ---------------- 1: per-edge dot + segment max (one wave32 per edge) -------
__global__ void k_edge_dot(const float* __restrict__ ent,
                           const float* __restrict__ rel,
                           const int* __restrict__ rel_id,
                           const int* __restrict__ dst,
                           float* __restrict__ s, unsigned* __restrict__ smax) {
    int wave = (blockIdx.x * blockDim.x + threadIdx.x) >> 5;
    int lane = threadIdx.x & 31;
    if (wave >= NE) return;
    int rid = rel_id[wave];
    int d   = dst[wave];
    const float4* ep = (const float4*)(rel + (size_t)rid * HD);
    const float4* vp = (const float4*)(ent + (size_t)d * HD);
    float4 a = ep[lane];
    float4 b = vp[lane];
    float p = a.x * b.x + a.y * b.y + a.z * b.z + a.w * b.w;
    p += __shfl_xor(p, 16);
    p += __shfl_xor(p, 8);
    p += __shfl_xor(p, 4);
    p += __shfl_xor(p, 2);
    p += __shfl_xor(p, 1);
    if (lane == 0) {
        s[wave] = p;
        atomicMax(&smax[d], enc_f32(p));
    }
}

// ---------------- 2: exp(s - smax[dst]) + segment sum (one thread per edge) -
__global__ void k_edge_exp(const float* __restrict__ s,
                           const int* __restrict__ dst,
                           const unsigned* __restrict__ smax,
                           float* __restrict__ ex, float* __restrict__ denom) {
    int e = blockIdx.x * blockDim.x + threadIdx.x;
    if (e >= NE) return;
    int d = dst[e];
    float v = expf(s[e] - dec_f32(smax[d]));
    ex[e] = v;
    atomicAdd(&denom[d], v);
}

// ---------------- 3: weighted scatter of edge embeddings (wave per edge) ----
__global__ void k_edge_scatter(const float* __restrict__ rel,
                               const int* __restrict__ rel_id,
                               const int* __restrict__ dst,
                               const float* __restrict__ ex,
                               const float* __restrict__ denom,
                               float* __restrict__ neigh) {
    int wave = (blockIdx.x * blockDim.x + threadIdx.x) >> 5;
    int lane = threadIdx.x & 31;
    if (wave >= NE) return;
    int rid = rel_id[wave];
    int d   = dst[wave];
    float coef = ex[wave] / denom[d];
    const float4* ep = (const float4*)(rel + (size_t)rid * HD);
    float4 r = ep[lane];
    float* np = neigh + (size_t)d * HD + lane * 4;
    atomicAdd(np + 0, r.x * coef);
    atomicAdd(np + 1, r.y * coef);
    atomicAdd(np + 2, r.z * coef);
    atomicAdd(np + 3, r.w * coef);
}

// ---------------- 4: h = neigh @ W via V_WMMA_F32_16X16X4_F32 ---------------
// Block = 256 threads = 8 waves; blockIdx = row tile (16 rows), wave = col tile.
__global__ void k_gemm_wmma(const float* __restrict__ A,   // [NN, HD] neigh
                            const float* __restrict__ B,   // [HD, HD] weights
                            float* __restrict__ Hout) {    // [NN, HD]
    int colTile = threadIdx.x >> 5;       // 0..7
    int lane    = threadIdx.x & 31;
    int rowTile = blockIdx.x;             // 0..2499
    int m     = lane & 15;
    int khalf = lane >> 4;

    const float* Arow = A + ((size_t)rowTile * 16 + m) * HD;     // A[m][*]
    const float* Bcol = B + colTile * 16 + m;                    // B[*][n], n=m

    v8f c = {};
#pragma unroll
    for (int kk = 0; kk < 32; ++kk) {
        int k = kk * 4 + khalf * 2;
        v2f a, b;
        a.x = Arow[k];
        a.y = Arow[k + 1];
        b.x = Bcol[(size_t)k * HD];
        b.y = Bcol[(size_t)(k + 1) * HD];
        c = __builtin_amdgcn_wmma_f32_16x16x4_f32(
                false, a, false, b, (short)0, c, false, false);
    }
    // C/D layout: VGPR g -> row (g + 8*khalf), col (lane&15)
    float* orow = Hout + ((size_t)rowTile * 16 + 8 * khalf) * HD
                       + colTile * 16 + m;
#pragma unroll
    for (int g = 0; g < 8; ++g)
        orow[(size_t)g * HD] = c[g];
}

// ---------------- 5: per-column sum / sumsq over the 40000-row batch --------
__global__ void k_colstats(const float* __restrict__ h,
                           float* __restrict__ colsum,
                           float* __restrict__ colsumsq, int rowsPerBlock) {
    int c  = threadIdx.x;                 // 128 threads = 128 columns
    int r0 = blockIdx.x * rowsPerBlock;
    int r1 = r0 + rowsPerBlock;
    if (r1 > NN) r1 = NN;
    float sm = 0.0f, sq = 0.0f;
    for (int r = r0; r < r1; ++r) {
        float v = h[(size_t)r * HD + c];
        sm += v;
        sq += v * v;
    }
    atomicAdd(&colsum[c], sm);
    atomicAdd(&colsumsq[c], sq);
}

// ---------------- 6: fold mean/var/gamma/beta into scale/shift --------------
__global__ void k_bnparam(const float* __restrict__ colsum,
                          const float* __restrict__ colsumsq,
                          const float* __restrict__ gamma,
                          const float* __restrict__ beta,
                          float* __restrict__ scale, float* __restrict__ shift) {
    int c = threadIdx.x;
    float invN = 1.0f / (float)NN;
    float mean = colsum[c] * invN;
    float var  = colsumsq[c] * invN - mean * mean;
    float sc   = gamma[c] * rsqrtf(var + BN_EPS);
    scale[c] = sc;
    shift[c] = beta[c] - mean * sc;
}

// ---------------- 7: BN + tanh epilogue, in-place on d_out ------------------
__global__ void k_bn_tanh(float* __restrict__ h,
                          const float* __restrict__ scale,
                          const float* __restrict__ shift) {
    int i = blockIdx.x * blockDim.x + threadIdx.x;
    if (i >= NN * HD) return;
    int c = i & (HD - 1);
    h[i] = tanhf(h[i] * scale[c] + shift[c]);
}

extern "C" void kernel_launch(void* const* d_in, const int* in_sizes, int n_in,
                              void* d_out, int out_size, void* d_ws, size_t ws_size,
                              hipStream_t stream) {
    const float* ent_emb  = (const float*)d_in[0];   // [NN, HD]
    const float* rel_emb  = (const float*)d_in[1];   // [474, HD]
    const float* neigh_w  = (const float*)d_in[2];   // [HD, HD]
    const float* bn_gamma = (const float*)d_in[3];   // [HD]
    const float* bn_beta  = (const float*)d_in[4];   // [HD]
    const int*   rel_id   = (const int*)d_in[5];     // [NE]
    const int*   dst      = (const int*)d_in[6];     // [NE]
    float* out = (float*)d_out;                      // [NN, HD]

    char* ws = (char*)d_ws;
    float*    s_buf = (float*)ws;              ws += (size_t)NE * 4;
    float*    ex    = (float*)ws;              ws += (size_t)NE * 4;
    unsigned* smax  = (unsigned*)ws;           ws += (size_t)NN * 4;
    float*    denom = (float*)ws;              ws += (size_t)NN * 4;
    float*    neigh = (float*)ws;              ws += (size_t)NN * HD * 4;
    float*    colsum   = (float*)ws;           ws += 128 * 4;
    float*    colsumsq = (float*)ws;           ws += 128 * 4;
    float*    scale    = (float*)ws;           ws += 128 * 4;
    float*    shift    = (float*)ws;           ws += 128 * 4;

    const int T = 256;

    // 0: init (covers NN*HD elements, largest array)
    k_init<<<(NN * HD + T - 1) / T, T, 0, stream>>>(smax, denom, neigh, colsum);

    // 1: per-edge dot + segment max (wave per edge -> NE*32 threads)
    {
        size_t threads = (size_t)NE * 32;
        k_edge_dot<<<(unsigned)((threads + T - 1) / T), T, 0, stream>>>(
            ent_emb, rel_emb, rel_id, dst, s_buf, smax);
    }
    // 2: exp + segment sum
    k_edge_exp<<<(NE + T - 1) / T, T, 0, stream>>>(s_buf, dst, smax, ex, denom);

    // 3: weighted scatter (wave per edge)
    {
        size_t threads = (size_t)NE * 32;
        k_edge_scatter<<<(unsigned)((threads + T - 1) / T), T, 0, stream>>>(
            rel_emb, rel_id, dst, ex, denom, neigh);
    }
    // 4: WMMA GEMM -> d_out holds h
    k_gemm_wmma<<<NN / 16, 256, 0, stream>>>(neigh, neigh_w, out);

    // 5: column stats (250 blocks x 160 rows)
    k_colstats<<<250, 128, 0, stream>>>(out, colsum, colsumsq, 160);

    // 6: BN params
    k_bnparam<<<1, 128, 0, stream>>>(colsum, colsumsq, bn_gamma, bn_beta, scale, shift);

    // 7: BN + tanh, in place
    k_bn_tanh<<<(NN * HD + T - 1) / T, T, 0, stream>>>(out, scale, shift);
}